// EffectiveLossFunction_65833258713642
// MI455X (gfx1250) — compile-verified
//
#include <hip/hip_runtime.h>
#include <stdint.h>

#ifndef __has_builtin
#define __has_builtin(x) 0
#endif

#define EPS_F    1e-5f
#define ONE_MEPS 0.99999f
#define EXPEPS   1.0000100000500002f   /* exp(1e-5) */

#if defined(__AMDGCN__) && __has_builtin(__builtin_amdgcn_global_load_async_to_lds_b128)
#define USE_ASYNC_LDS 1
#else
#define USE_ASYNC_LDS 0
#endif

#if defined(__AMDGCN__)
#if __has_builtin(__builtin_amdgcn_s_wait_asynccnt)
#define WAIT_ASYNC(n) __builtin_amdgcn_s_wait_asynccnt(n)
#else
#define WAIT_ASYNC(n) asm volatile("s_wait_asynccnt %0" ::"i"(n))
#endif
#define WAIT_DS0() asm volatile("s_wait_dscnt 0" ::: "memory")
#else
#define WAIT_ASYNC(n)
#define WAIT_DS0()
#endif

// per-component update:  acc += T*o ;  T *= (1-o)
#define RAY_STEP(c)                                         \
  do {                                                      \
    float o = fminf(fmaxf(v.c, EPS_F), ONE_MEPS);           \
    acc.c = fmaf(T.c, o, acc.c);                            \
    T.c   = fmaf(-o, T.c, T.c);                             \
  } while (0)

// first depth:  acc = e^eps * o ;  T = 1-o
#define RAY_INIT(c)                                         \
  do {                                                      \
    float o = fminf(fmaxf(v.c, EPS_F), ONE_MEPS);           \
    acc.c = EXPEPS * o;                                     \
    T.c   = 1.0f - o;                                       \
  } while (0)

namespace {
constexpr int DK     = 128;        // depth
constexpr int WD     = 128;        // width (= height)
constexpr int HWC    = 128 * 128;  // one depth slice per batch
constexpr int TPB    = 256;        // threads per block (8 waves, wave32)
constexpr int RPT    = 4;          // rays per thread (float4 along w)
constexpr int STAGES = 16;         // async pipeline depth (16 MB in flight grid-wide)
}  // namespace

__global__ __launch_bounds__(TPB) void effloss_raycast_kernel(
    const float* __restrict__ vox, float* __restrict__ out) {
  const int tid  = threadIdx.x;
  const int ray0 = blockIdx.x * (TPB * RPT) + tid * RPT;  // 4 consecutive rays
  const int b    = ray0 >> 14;                            // / (128*128)
  const int hw   = ray0 & (HWC - 1);
  const float* g0 = vox + (size_t)b * DK * HWC + hw;      // depth-0 address

  float4 T   = {1.f, 1.f, 1.f, 1.f};
  float4 acc = {0.f, 0.f, 0.f, 0.f};

#if USE_ASYNC_LDS
  typedef int v4i __attribute__((vector_size(16)));
  typedef __attribute__((address_space(1))) v4i gv4i;  // global int4
  typedef __attribute__((address_space(3))) v4i lv4i;  // LDS int4
  __shared__ float4 lbuf[STAGES * TPB];  // 64 KB; thread-private slots
  float4* slot0 = &lbuf[tid];

  // prologue: launch STAGES depth slices into LDS (ASYNCcnt-tracked)
#pragma unroll
  for (int s = 0; s < STAGES; ++s) {
    __builtin_amdgcn_global_load_async_to_lds_b128(
        (gv4i*)(uintptr_t)(g0 + (size_t)s * HWC),
        (lv4i*)(uintptr_t)(slot0 + s * TPB), 0, 0);
  }

  // d = 0 (carries the exp(EPS) factor of the reference's eps slab)
  WAIT_ASYNC(STAGES - 1);  // oldest async load complete (in-order completion)
  {
    float4 v = slot0[0];
    RAY_INIT(x); RAY_INIT(y); RAY_INIT(z); RAY_INIT(w);
    WAIT_DS0();  // LDS read data in VGPRs before slot is overwritten
    __builtin_amdgcn_global_load_async_to_lds_b128(
        (gv4i*)(uintptr_t)(g0 + (size_t)STAGES * HWC),
        (lv4i*)(uintptr_t)(slot0 + 0), 0, 0);
  }

  // steady state: consume depth d, refill same slot with depth d+STAGES
  for (int d = 1; d < DK - STAGES; ++d) {
    WAIT_ASYNC(STAGES - 1);
    const int s = d & (STAGES - 1);
    float4 v = slot0[s * TPB];
    RAY_STEP(x); RAY_STEP(y); RAY_STEP(z); RAY_STEP(w);
    WAIT_DS0();
    __builtin_amdgcn_global_load_async_to_lds_b128(
        (gv4i*)(uintptr_t)(g0 + (size_t)(d + STAGES) * HWC),
        (lv4i*)(uintptr_t)(slot0 + s * TPB), 0, 0);
  }

  // epilogue: everything issued; drain
  WAIT_ASYNC(0);
#pragma unroll
  for (int d = DK - STAGES; d < DK; ++d) {
    float4 v = slot0[(d & (STAGES - 1)) * TPB];
    RAY_STEP(x); RAY_STEP(y); RAY_STEP(z); RAY_STEP(w);
  }
#else
  // Fallback: register-pipelined coalesced b128 global loads
  {
    float4 v = *(const float4*)g0;
    RAY_INIT(x); RAY_INIT(y); RAY_INIT(z); RAY_INIT(w);
  }
  float4 nxt = *(const float4*)(g0 + (size_t)HWC);
#pragma unroll 4
  for (int d = 1; d < DK; ++d) {
    float4 v = nxt;
    if (d + 1 < DK) nxt = *(const float4*)(g0 + (size_t)(d + 1) * HWC);
    RAY_STEP(x); RAY_STEP(y); RAY_STEP(z); RAY_STEP(w);
  }
#endif

  // project done; apply vertical flip of H on store
  const int h = hw >> 7;
  const int w = hw & (WD - 1);
  float4* op = (float4*)(out + (size_t)b * HWC + (size_t)(WD - 1 - h) * WD + w);
  *op = acc;
}

extern "C" void kernel_launch(void* const* d_in, const int* in_sizes, int n_in,
                              void* d_out, int out_size, void* d_ws, size_t ws_size,
                              hipStream_t stream) {
  (void)n_in; (void)out_size; (void)d_ws; (void)ws_size;
  const float* vox = (const float*)d_in[0];
  float* out = (float*)d_out;
  const int total  = in_sizes[0];          // B * 128^3
  const int rays   = total / DK;           // B * H * W
  const int blocks = rays / (TPB * RPT);   // 256 for B=16
  hipLaunchKernelGGL(effloss_raycast_kernel, dim3(blocks), dim3(TPB), 0, stream,
                     vox, out);
}